// KDEClassifier_39273180954694
// MI455X (gfx1250) — compile-verified
//
#include <hip/hip_runtime.h>
#include <hip/hip_bf16.h>

typedef __attribute__((ext_vector_type(2))) float v2f;
typedef __attribute__((ext_vector_type(4))) float v4f;
typedef __attribute__((ext_vector_type(8))) float v8f;

#define QN     8192
#define DFEAT  384
#define DP     16
#define NDATA  40000
#define NTILES (NDATA / DP)          // 2500
#define WAVES_PER_BLOCK 8
#define CHUNK_Y 4
#define LOG2E  1.4426950408889634f

// ---------------------------------------------------------------------------
// Stage 1: h = ((features - mean) @ SV / max_length) @ bw
//   store h_scaled = h * log2(e)          (WMMA A operand, exp2 domain)
//   store hn_scaled = -0.5*log2(e)*|h|^2
// ---------------------------------------------------------------------------
__global__ void __launch_bounds__(256) prep_h(
    const float* __restrict__ feat, const float* __restrict__ mean,
    const float* __restrict__ sv,   const float* __restrict__ maxlen,
    const float* __restrict__ bw,   float* __restrict__ h,
    float* __restrict__ hn)
{
    __shared__ float s_mean[DFEAT];
    __shared__ float s_sv[DFEAT * DP];
    __shared__ float s_bw[DP * DP];
    const int tid = threadIdx.x;
    for (int i = tid; i < DFEAT; i += blockDim.x)      s_mean[i] = mean[i];
    for (int i = tid; i < DFEAT * DP; i += blockDim.x) s_sv[i]   = sv[i];
    for (int i = tid; i < DP * DP; i += blockDim.x)    s_bw[i]   = bw[i];
    __syncthreads();

    const int q = blockIdx.x * blockDim.x + tid;
    if (q >= QN) return;

    const float inv_ml = 1.0f / maxlen[0];
    float g[DP];
#pragma unroll
    for (int d = 0; d < DP; ++d) g[d] = 0.0f;

    const float* frow = feat + (size_t)q * DFEAT;
    for (int k = 0; k < DFEAT; ++k) {
        const float x = frow[k] - s_mean[k];
#pragma unroll
        for (int d = 0; d < DP; ++d) g[d] = fmaf(x, s_sv[k * DP + d], g[d]);
    }
#pragma unroll
    for (int d = 0; d < DP; ++d) g[d] *= inv_ml;

    float nrm = 0.0f;
#pragma unroll
    for (int dp = 0; dp < DP; ++dp) {
        float v = 0.0f;
#pragma unroll
        for (int d = 0; d < DP; ++d) v = fmaf(g[d], s_bw[d * DP + dp], v);
        h[(size_t)q * DP + dp] = v * LOG2E;   // pre-scaled A operand
        nrm = fmaf(v, v, nrm);
    }
    hn[q] = -0.5f * LOG2E * nrm;
}

// ---------------------------------------------------------------------------
// Stage 2a: dsn_j = -0.5*log2(e)*|dataset_j|^2
// ---------------------------------------------------------------------------
__global__ void prep_dsn(const float* __restrict__ ds, float* __restrict__ dsn)
{
    const int j = blockIdx.x * blockDim.x + threadIdx.x;
    if (j >= NDATA) return;
    const float4* r = (const float4*)(ds + (size_t)j * DP);
    float s = 0.0f;
#pragma unroll
    for (int i = 0; i < 4; ++i) {
        float4 v = r[i];
        s += v.x * v.x + v.y * v.y + v.z * v.z + v.w * v.w;
    }
    dsn[j] = -0.5f * LOG2E * s;
}

// ---------------------------------------------------------------------------
// Stage 2b: repack dataset into per-lane WMMA B-fragment order.
// Tile t, lane l (hi=l>>4, l16=l&15) owns B elements
//   ds[t*16+l16][4k + 2*hi + {0,1}], k=0..3  -> 8 contiguous floats (32B)
// ---------------------------------------------------------------------------
__global__ void pack_b(const float* __restrict__ ds, float* __restrict__ pack)
{
    const int idx = blockIdx.x * blockDim.x + threadIdx.x;   // t*32 + lane
    if (idx >= NTILES * 32) return;
    const int t = idx >> 5, lane = idx & 31;
    const int hi = lane >> 4, l16 = lane & 15;
    const float* drow = ds + (size_t)(t * DP + l16) * DP + 2 * hi;
    float* p = pack + (size_t)idx * 8;
    p[0] = drow[0];  p[1] = drow[1];
    p[2] = drow[4];  p[3] = drow[5];
    p[4] = drow[8];  p[5] = drow[9];
    p[6] = drow[12]; p[7] = drow[13];
}

__global__ void zero_acc(float* __restrict__ a, int n)
{
    const int i = blockIdx.x * blockDim.x + threadIdx.x;
    if (i < n) a[i] = 0.0f;
}

// ---------------------------------------------------------------------------
// Stage 3: per 16-query tile, sweep dataset tiles with V_WMMA_F32_16X16X4_F32.
// WMMA C is pre-seeded with (hn'_q + dsn'_j) so the matrix pipe carries the
// bias; then one bare v_exp_f32 per element (underflow->0 is desired).
// ---------------------------------------------------------------------------
__global__ void __launch_bounds__(256) kde_main(
    const float* __restrict__ h,    const float* __restrict__ hn,
    const float* __restrict__ pack, const float* __restrict__ dsn,
    float* __restrict__ acc_g)
{
    __shared__ float sacc[DP];

    const int lane = threadIdx.x & 31;
    const int wave = __builtin_amdgcn_readfirstlane(threadIdx.x >> 5); // wave-uniform
    const int hi   = lane >> 4;
    const int l16  = lane & 15;
    const int q0   = blockIdx.x * DP;

    const int nchunks = CHUNK_Y * WAVES_PER_BLOCK;     // 32
    const int chunk   = blockIdx.y * WAVES_PER_BLOCK + wave;
    const int tpc     = (NTILES + nchunks - 1) / nchunks;
    const int t0      = chunk * tpc;
    const int t1      = min(t0 + tpc, NTILES);

    if (threadIdx.x < DP) sacc[threadIdx.x] = 0.0f;
    __syncthreads();

    // A fragments (constant across the j-loop); 32-bit A 16x4 layout:
    // lanes 0-15 hold K={k0,k0+1}, lanes 16-31 hold K={k0+2,k0+3}
    const float* hrow = h + (size_t)(q0 + l16) * DP + 2 * hi;
    const v2f a0 = *(const v2f*)(hrow + 0);
    const v2f a1 = *(const v2f*)(hrow + 4);
    const v2f a2 = *(const v2f*)(hrow + 8);
    const v2f a3 = *(const v2f*)(hrow + 12);

    v8f hnv;
#pragma unroll
    for (int v = 0; v < 8; ++v) hnv[v] = hn[q0 + 8 * hi + v];

    float acc[8];
#pragma unroll
    for (int v = 0; v < 8; ++v) acc[v] = 0.0f;

    const float* pbase = pack + (size_t)lane * 8;

#pragma unroll 2
    for (int t = t0; t < t1; ++t) {
        const float* p = pbase + (size_t)t * 256;   // 32 lanes * 8 floats per tile
        __builtin_prefetch(p + 256, 0, 3);          // next tile, near-cache

        const v4f p0 = *(const v4f*)(p + 0);
        const v4f p1 = *(const v4f*)(p + 4);
        const float dn = dsn[t * DP + l16];

        const v2f b0 = {p0.x, p0.y};
        const v2f b1 = {p0.z, p0.w};
        const v2f b2 = {p1.x, p1.y};
        const v2f b3 = {p1.z, p1.w};

        // Seed C with the exp2-domain bias: c[v] = hn' + dsn'
        v8f c;
#pragma unroll
        for (int v = 0; v < 8; ++v) c[v] = hnv[v] + dn;

        c = __builtin_amdgcn_wmma_f32_16x16x4_f32(false, a0, false, b0, (short)0, c, false, false);
        c = __builtin_amdgcn_wmma_f32_16x16x4_f32(false, a1, false, b1, (short)0, c, false, false);
        c = __builtin_amdgcn_wmma_f32_16x16x4_f32(false, a2, false, b2, (short)0, c, false, false);
        c = __builtin_amdgcn_wmma_f32_16x16x4_f32(false, a3, false, b3, (short)0, c, false, false);

#pragma unroll
        for (int v = 0; v < 8; ++v)
            acc[v] += __builtin_amdgcn_exp2f(c[v]);   // bare v_exp_f32; underflow -> 0
    }

    // Reduce over the 16 N-lanes of each half (xor masks < 16 keep halves separate)
#pragma unroll
    for (int m = 8; m >= 1; m >>= 1) {
#pragma unroll
        for (int v = 0; v < 8; ++v) acc[v] += __shfl_xor(acc[v], m, 32);
    }

    if (l16 == 0) {
#pragma unroll
        for (int v = 0; v < 8; ++v) atomicAdd(&sacc[8 * hi + v], acc[v]);
    }
    __syncthreads();
    if (threadIdx.x < DP) atomicAdd(&acc_g[q0 + threadIdx.x], sacc[threadIdx.x]);
}

// ---------------------------------------------------------------------------
// Stage 4: probabilities = 1 / (1 + exp(0.05 * (log(mean*norm) - 12)))
// ---------------------------------------------------------------------------
__global__ void finalize(const float* __restrict__ acc,
                         const float* __restrict__ normp,
                         float* __restrict__ out)
{
    const int q = blockIdx.x * blockDim.x + threadIdx.x;
    if (q >= QN) return;
    const float est   = acc[q] * (normp[0] / (float)NDATA);
    const float score = logf(est);               // est==0 -> -inf -> out==1 (matches f32 ref)
    out[q] = 1.0f / (1.0f + expf(0.05f * (score - 12.0f)));
}

extern "C" void kernel_launch(void* const* d_in, const int* in_sizes, int n_in,
                              void* d_out, int out_size, void* d_ws, size_t ws_size,
                              hipStream_t stream)
{
    const float* feat = (const float*)d_in[0];
    const float* mean = (const float*)d_in[1];
    const float* sv   = (const float*)d_in[2];
    const float* ml   = (const float*)d_in[3];
    const float* bw   = (const float*)d_in[4];
    const float* ds   = (const float*)d_in[5];
    const float* nrm  = (const float*)d_in[6];
    float* out = (float*)d_out;

    float* h    = (float*)d_ws;                // QN*DP
    float* hn   = h + (size_t)QN * DP;         // QN
    float* dsn  = hn + QN;                     // NDATA
    float* acc  = dsn + NDATA;                 // QN
    float* pack = acc + QN;                    // NTILES*32*8 = 640000 floats

    prep_h<<<QN / 256, 256, 0, stream>>>(feat, mean, sv, ml, bw, h, hn);
    prep_dsn<<<(NDATA + 255) / 256, 256, 0, stream>>>(ds, dsn);
    pack_b<<<(NTILES * 32 + 255) / 256, 256, 0, stream>>>(ds, pack);
    zero_acc<<<(QN + 255) / 256, 256, 0, stream>>>(acc, QN);

    dim3 grid(QN / DP, CHUNK_Y);
    kde_main<<<grid, 256, 0, stream>>>(h, hn, pack, dsn, acc);

    finalize<<<(QN + 255) / 256, 256, 0, stream>>>(acc, nrm, out);
}